// RejectionSampler_36009005809787
// MI455X (gfx1250) — compile-verified
//
#include <hip/hip_runtime.h>
#include <math.h>

// Problem constants (match reference): B=128 requests, S=8 spec tokens, V=32000 vocab.
#define NT 256
constexpr int kB = 128;
constexpr int kS = 8;
constexpr int kV = 32000;
constexpr int kT = kB * kS;            // 1024 tokens
constexpr int kPLACEHOLDER = -1;
constexpr int kChunk = 4 * NT;         // 1024 floats moved per block per iter
constexpr int kIters = kV / kChunk;    // 31 uniform iterations
constexpr int kTail  = kIters * kChunk;       // 31744
constexpr int kTailN = (kV - kTail) / 4;      // 64 float4s in the tail

#if defined(__has_builtin)
#  if __has_builtin(__builtin_amdgcn_global_load_async_to_lds_b128)
#    define USE_ASYNC_LDS 1
#  endif
#endif

// 16-byte vector type matching the async-to-LDS builtin's expected pointee.
typedef int v4i __attribute__((ext_vector_type(4)));
typedef __attribute__((address_space(1))) v4i* gas_v4i_ptr;
typedef __attribute__((address_space(3))) v4i* lds_v4i_ptr;

__device__ __forceinline__ void async_ld16(const float* g, float* l) {
#ifdef USE_ASYNC_LDS
    __builtin_amdgcn_global_load_async_to_lds_b128((gas_v4i_ptr)g, (lds_v4i_ptr)l, 0, 0);
#else
    *(float4*)l = *(const float4*)g;
#endif
}

// One workgroup per token. 8 waves (wave32) stream the 125KB logits row into LDS
// via CDNA5 async global->LDS DMA, do an LDS-resident softmax, then a fused
// residual-argmax over draft/q streamed from global (prefetched ahead of use).
__global__ __launch_bounds__(NT) void rs_main_kernel(
    const float* __restrict__ logits,       // [T, V]
    const float* __restrict__ draft,        // [T, V]
    const int*   __restrict__ draft_ids,    // [T]
    const float* __restrict__ temperature,  // [B]
    const float* __restrict__ uniform,      // [T]
    const float* __restrict__ q,            // [B, V]
    int* __restrict__ ws_accept,            // [T]
    int* __restrict__ ws_tok)               // [T]
{
    __shared__ __align__(16) float e[kV];   // 125 KB of the 320 KB/WGP LDS
    __shared__ float redf[8];               // one slot per wave
    __shared__ int   redi[8];

    const int t    = blockIdx.x;            // token index
    const int req  = t >> 3;                // request index (S == 8)
    const int tid  = threadIdx.x;
    const int lane = tid & 31;
    const int wid  = tid >> 5;

    const float invT = 1.0f / temperature[req];
    const float* __restrict__ lrow = logits + (size_t)t   * kV;
    const float* __restrict__ drow = draft  + (size_t)t   * kV;
    const float* __restrict__ qrow = q      + (size_t)req * kV;

    // ---- Stage logits row into LDS: CDNA5 async global->LDS DMA (ASYNCcnt) ----
    {
        int i = 4 * tid;
        #pragma unroll 4
        for (int it = 0; it < kIters; ++it, i += kChunk)
            async_ld16(lrow + i, &e[i]);
        if (tid < kTailN)
            async_ld16(lrow + kTail + 4 * tid, &e[kTail + 4 * tid]);
    }

    // ---- Prefetch draft/q rows (global_prefetch) while softmax runs on LDS ----
    for (int off = tid * 32; off < kV; off += NT * 32) {   // 128B cachelines
        __builtin_prefetch(drow + off, 0, 1);
        __builtin_prefetch(qrow + off, 0, 1);
    }

#ifdef USE_ASYNC_LDS
#  if __has_builtin(__builtin_amdgcn_s_wait_asynccnt)
    __builtin_amdgcn_s_wait_asynccnt(0);
#  else
    asm volatile("s_wait_asynccnt 0" ::: "memory");
#  endif
#endif
    __syncthreads();

    // ---- Phase 1: row max (raw logits; temp > 0 so scaling commutes with max) ----
    float m = -INFINITY;
    {
        int i = 4 * tid;
        #pragma unroll 4
        for (int it = 0; it < kIters; ++it, i += kChunk) {
            float4 x = *(float4*)(&e[i]);
            m = fmaxf(m, fmaxf(fmaxf(x.x, x.y), fmaxf(x.z, x.w)));
        }
        if (tid < kTailN) {
            float4 x = *(float4*)(&e[kTail + 4 * tid]);
            m = fmaxf(m, fmaxf(fmaxf(x.x, x.y), fmaxf(x.z, x.w)));
        }
    }
    #pragma unroll
    for (int o = 16; o > 0; o >>= 1) m = fmaxf(m, __shfl_xor(m, o, 32));
    if (lane == 0) redf[wid] = m;
    __syncthreads();
    float M = redf[0];
    #pragma unroll
    for (int w = 1; w < 8; ++w) M = fmaxf(M, redf[w]);
    __syncthreads();

    // ---- Phase 2: e = exp((l - M) * invT) written back in place; sum reduce ----
    float s = 0.0f;
    {
        int i = 4 * tid;
        #pragma unroll 2
        for (int it = 0; it < kIters; ++it, i += kChunk) {
            float4 x = *(float4*)(&e[i]);
            float4 ev;
            ev.x = __expf((x.x - M) * invT);
            ev.y = __expf((x.y - M) * invT);
            ev.z = __expf((x.z - M) * invT);
            ev.w = __expf((x.w - M) * invT);
            s += (ev.x + ev.y) + (ev.z + ev.w);
            *(float4*)(&e[i]) = ev;
        }
        if (tid < kTailN) {
            const int i2 = kTail + 4 * tid;
            float4 x = *(float4*)(&e[i2]);
            float4 ev;
            ev.x = __expf((x.x - M) * invT);
            ev.y = __expf((x.y - M) * invT);
            ev.z = __expf((x.z - M) * invT);
            ev.w = __expf((x.w - M) * invT);
            s += (ev.x + ev.y) + (ev.z + ev.w);
            *(float4*)(&e[i2]) = ev;
        }
    }
    #pragma unroll
    for (int o = 16; o > 0; o >>= 1) s += __shfl_xor(s, o, 32);
    if (lane == 0) redf[wid] = s;
    __syncthreads();
    float Ssum = redf[0];
    #pragma unroll
    for (int w = 1; w < 8; ++w) Ssum += redf[w];
    __syncthreads();

    // ---- Phase 3: argmax_v max(e - Ssum*d, 0)/q  (== argmax of max(p-d,0)/q) ----
    float best = -1.0f;
    int   bi   = 0x7fffffff;
    {
        int i = 4 * tid;
        #pragma unroll 2
        for (int it = 0; it < kIters; ++it, i += kChunk) {
            float4 ev = *(float4*)(&e[i]);
            float4 d4 = *(const float4*)(&drow[i]);
            float4 q4 = *(const float4*)(&qrow[i]);
            float r0 = fmaxf(ev.x - Ssum * d4.x, 0.0f) / q4.x;
            float r1 = fmaxf(ev.y - Ssum * d4.y, 0.0f) / q4.y;
            float r2 = fmaxf(ev.z - Ssum * d4.z, 0.0f) / q4.z;
            float r3 = fmaxf(ev.w - Ssum * d4.w, 0.0f) / q4.w;
            if (r0 > best) { best = r0; bi = i;     }   // strict > keeps first index
            if (r1 > best) { best = r1; bi = i + 1; }
            if (r2 > best) { best = r2; bi = i + 2; }
            if (r3 > best) { best = r3; bi = i + 3; }
        }
        if (tid < kTailN) {
            const int i2 = kTail + 4 * tid;
            float4 ev = *(float4*)(&e[i2]);
            float4 d4 = *(const float4*)(&drow[i2]);
            float4 q4 = *(const float4*)(&qrow[i2]);
            float r0 = fmaxf(ev.x - Ssum * d4.x, 0.0f) / q4.x;
            float r1 = fmaxf(ev.y - Ssum * d4.y, 0.0f) / q4.y;
            float r2 = fmaxf(ev.z - Ssum * d4.z, 0.0f) / q4.z;
            float r3 = fmaxf(ev.w - Ssum * d4.w, 0.0f) / q4.w;
            if (r0 > best) { best = r0; bi = i2;     }
            if (r1 > best) { best = r1; bi = i2 + 1; }
            if (r2 > best) { best = r2; bi = i2 + 2; }
            if (r3 > best) { best = r3; bi = i2 + 3; }
        }
    }
    #pragma unroll
    for (int o = 16; o > 0; o >>= 1) {
        float v2 = __shfl_xor(best, o, 32);
        int   i2 = __shfl_xor(bi,   o, 32);
        if (v2 > best || (v2 == best && i2 < bi)) { best = v2; bi = i2; }
    }
    if (lane == 0) { redf[wid] = best; redi[wid] = bi; }
    __syncthreads();

    // ---- Phase 4: final cross-wave argmax + accept test (single lane) ----
    if (tid == 0) {
        float bv = redf[0];
        int   bx = redi[0];
        #pragma unroll
        for (int w = 1; w < 8; ++w) {
            float v2 = redf[w];
            int   i2 = redi[w];
            if (v2 > bv || (v2 == bv && i2 < bx)) { bv = v2; bx = i2; }
        }
        const int   dtok  = draft_ids[t];
        const float p_tgt = e[dtok] / Ssum;
        const float p_drf = drow[dtok];
        const int accept  = (p_drf > 0.0f) && (p_tgt >= uniform[t] * p_drf);
        ws_accept[t] = accept;
        ws_tok[t]    = accept ? dtok : bx;
    }
}

// Sequential accept-until-first-reject scan per request; bonus token if all accepted.
__global__ __launch_bounds__(kB) void rs_final_kernel(
    const int* __restrict__ ws_accept,   // [T]
    const int* __restrict__ ws_tok,      // [T]
    const int* __restrict__ bonus_ids,   // [B]
    int* __restrict__ out)               // [B, S+1]
{
    const int b = threadIdx.x;
    if (b >= kB) return;
    int alive = 1;
    #pragma unroll
    for (int p = 0; p < kS; ++p) {
        const int idx = b * kS + p;
        out[b * (kS + 1) + p] = alive ? ws_tok[idx] : kPLACEHOLDER;
        alive &= ws_accept[idx];
    }
    out[b * (kS + 1) + kS] = alive ? bonus_ids[b] : kPLACEHOLDER;
}

extern "C" void kernel_launch(void* const* d_in, const int* in_sizes, int n_in,
                              void* d_out, int out_size, void* d_ws, size_t ws_size,
                              hipStream_t stream) {
    // setup_inputs() order:
    const float* logits      = (const float*)d_in[0];  // target_logits [T,V]
    const float* draft       = (const float*)d_in[1];  // draft_probs   [T,V]
    const int*   draft_ids   = (const int*)  d_in[2];  // [T]
    const int*   bonus_ids   = (const int*)  d_in[3];  // [B]
    const float* temperature = (const float*)d_in[4];  // [B]
    const float* uniform     = (const float*)d_in[5];  // [T]
    const float* q           = (const float*)d_in[6];  // [B,V]
    // d_in[7] = cu_num_draft_tokens (uniform raggedness; unused)

    int* ws_accept = (int*)d_ws;          // [T]
    int* ws_tok    = ws_accept + kT;      // [T]

    rs_main_kernel<<<kT, NT, 0, stream>>>(logits, draft, draft_ids, temperature,
                                          uniform, q, ws_accept, ws_tok);
    rs_final_kernel<<<1, kB, 0, stream>>>(ws_accept, ws_tok, bonus_ids, (int*)d_out);
}